// GaussianMixtureField_24627342475667
// MI455X (gfx1250) — compile-verified
//
#include <hip/hip_runtime.h>
#include <hip/hip_bf16.h>

typedef __attribute__((ext_vector_type(2))) float v2f;
typedef __attribute__((ext_vector_type(8))) float v8f;

#define LOG2E 1.4426950408889634f
#define NPT 4                 // 16-point tiles per wave
#define WAVES_PER_BLOCK 8     // 256 threads

// ---------------------------------------------------------------------------
// Kernel 1: per-Gaussian coefficient precompute (K threads).
// Builds fused 12-float coefficient vector g[k] such that the per-(point,
// gaussian) contribution is exp2( f(x) . g[k] ), with f(x) =
// [x0^2, x1^2, x2^2, x0x1, x0x2, x1x2, x0, x1, x2, 1, 0, 0].
// -0.5, log2(e) and log(amp) are all folded into g so the hot epilogue is a
// single v_exp_f32 per matrix element.
// ---------------------------------------------------------------------------
__global__ void gmf_precompute(const float* __restrict__ means,
                               const float* __restrict__ log_scales,
                               const float* __restrict__ quats,
                               const float* __restrict__ log_amps,
                               float* __restrict__ g, int K) {
    int k = blockIdx.x * blockDim.x + threadIdx.x;
    if (k >= K) return;

    // normalized quaternion -> rotation matrix
    float qw = quats[k * 4 + 0], qx = quats[k * 4 + 1];
    float qy = quats[k * 4 + 2], qz = quats[k * 4 + 3];
    float qn = rsqrtf(qw * qw + qx * qx + qy * qy + qz * qz);
    qw *= qn; qx *= qn; qy *= qn; qz *= qn;
    float R00 = 1.f - 2.f * (qy * qy + qz * qz);
    float R01 = 2.f * (qx * qy - qw * qz);
    float R02 = 2.f * (qx * qz + qw * qy);
    float R10 = 2.f * (qx * qy + qw * qz);
    float R11 = 1.f - 2.f * (qx * qx + qz * qz);
    float R12 = 2.f * (qy * qz - qw * qx);
    float R20 = 2.f * (qx * qz - qw * qy);
    float R21 = 2.f * (qy * qz + qw * qx);
    float R22 = 1.f - 2.f * (qx * qx + qy * qy);

    float s0 = fminf(fmaxf(expf(log_scales[k * 3 + 0]), 1e-5f), 100.f);
    float s1 = fminf(fmaxf(expf(log_scales[k * 3 + 1]), 1e-5f), 100.f);
    float s2 = fminf(fmaxf(expf(log_scales[k * 3 + 2]), 1e-5f), 100.f);
    float v0 = s0 * s0, v1 = s1 * s1, v2 = s2 * s2;

    // Sigma = R diag(v) R^T + 1e-5 I  (symmetric)
    float S00 = R00 * R00 * v0 + R01 * R01 * v1 + R02 * R02 * v2 + 1e-5f;
    float S11 = R10 * R10 * v0 + R11 * R11 * v1 + R12 * R12 * v2 + 1e-5f;
    float S22 = R20 * R20 * v0 + R21 * R21 * v1 + R22 * R22 * v2 + 1e-5f;
    float S01 = R00 * R10 * v0 + R01 * R11 * v1 + R02 * R12 * v2;
    float S02 = R00 * R20 * v0 + R01 * R21 * v1 + R02 * R22 * v2;
    float S12 = R10 * R20 * v0 + R11 * R21 * v1 + R12 * R22 * v2;

    // symmetric 3x3 inverse via cofactors
    float C00 = S11 * S22 - S12 * S12;
    float C01 = S02 * S12 - S01 * S22;
    float C02 = S01 * S12 - S02 * S11;
    float C11 = S00 * S22 - S02 * S02;
    float C12 = S01 * S02 - S00 * S12;
    float C22 = S00 * S11 - S01 * S01;
    float det = S00 * C00 + S01 * C01 + S02 * C02;
    float id = 1.f / det;
    float A00 = C00 * id, A01 = C01 * id, A02 = C02 * id;
    float A11 = C11 * id, A12 = C12 * id, A22 = C22 * id;

    float m0 = means[k * 3 + 0], m1 = means[k * 3 + 1], m2 = means[k * 3 + 2];
    float b0 = A00 * m0 + A01 * m1 + A02 * m2;
    float b1 = A01 * m0 + A11 * m1 + A12 * m2;
    float b2 = A02 * m0 + A12 * m1 + A22 * m2;
    float c = b0 * m0 + b1 * m1 + b2 * m2;
    float la = fminf(fmaxf(log_amps[k], -10.f), 6.f);

    const float sc = -0.5f * LOG2E;      // fold exp(-0.5*..) -> exp2
    float* gp = g + (size_t)k * 12;
    gp[0] = sc * A00;
    gp[1] = sc * A11;
    gp[2] = sc * A22;
    gp[3] = sc * 2.f * A01;
    gp[4] = sc * 2.f * A02;
    gp[5] = sc * 2.f * A12;
    gp[6] = sc * -2.f * b0;
    gp[7] = sc * -2.f * b1;
    gp[8] = sc * -2.f * b2;
    gp[9] = sc * c + la * LOG2E;         // constant term + folded amplitude
    gp[10] = 0.f;
    gp[11] = 0.f;
}

// ---------------------------------------------------------------------------
// Kernel 2: main evaluation. Each wave owns NPT 16-point tiles and sweeps all
// K/16 gaussian tiles. Per tile: 3x V_WMMA_F32_16X16X4_F32 (K = 12 features),
// epilogue = 8 independent v_exp_f32 + tree add per lane (no TRANS hazards).
// Coefficient loads are double-buffered one k-tile ahead.
//
// WMMA mapping: A = gaussians(16) x features(4/step)  [lane&15 -> gaussian M,
//   lane>>4 selects K pair per ISA 32-bit 16x4 A layout]
//               B = features(4/step) x points(16)     [lane&15 -> point N]
//               D[m,n]: lane holds 8 gaussian rows of its point column.
// ---------------------------------------------------------------------------
__global__ __launch_bounds__(256) void gmf_eval(const float* __restrict__ x,
                                                const float* __restrict__ g,
                                                float* __restrict__ out,
                                                int N, int K) {
    int lane = threadIdx.x & 31;
    int wid  = threadIdx.x >> 5;
    int col  = lane & 15;
    int h    = lane >> 4;  // which half-wave (selects K components)

    int gwave = blockIdx.x * WAVES_PER_BLOCK + wid;
    int nbase = gwave * (16 * NPT);

    // Per-lane B-operand values for each point tile. This lane only ever
    // supplies feature components {2h, 2h+1, 4+2h, 4+2h+1, 8+2h, 8+2h+1}.
    float bv[NPT][6];
    int pidx[NPT];
    for (int t = 0; t < NPT; ++t) {
        int p = nbase + t * 16 + col;
        pidx[t] = p;
        int pc = p < N ? p : N - 1;  // clamp loads (stores guarded below)
        float x0 = x[pc * 3 + 0];
        float x1 = x[pc * 3 + 1];
        float x2 = x[pc * 3 + 2];
        bv[t][0] = h ? x2 * x2 : x0 * x0;  // f2  : f0
        bv[t][1] = h ? x0 * x1 : x1 * x1;  // f3  : f1
        bv[t][2] = h ? x0      : x0 * x2;  // f6  : f4
        bv[t][3] = h ? x1      : x1 * x2;  // f7  : f5
        bv[t][4] = h ? 0.f     : x2;       // f10 : f8
        bv[t][5] = h ? 0.f     : 1.f;      // f11 : f9
    }

    float acc[NPT];
    for (int t = 0; t < NPT; ++t) acc[t] = 0.f;

    int ktiles = K >> 4;
    // per-lane base into coefficient table: gaussian (col) of tile 0, K-pair h
    const float* ga = g + (size_t)col * 12 + h * 2;

    // prime the pipeline: tile 0 coefficients
    v2f a0 = *(const v2f*)(ga + 0);
    v2f a1 = *(const v2f*)(ga + 4);
    v2f a2 = *(const v2f*)(ga + 8);

    for (int kt = 0; kt < ktiles; ++kt) {
        // prefetch next k-tile (clamped so last iteration stays in-bounds,
        // keeps control flow uniform for WMMA's EXEC==all-ones requirement)
        int ktn = kt + 1 < ktiles ? kt + 1 : ktiles - 1;
        const float* gn = ga + (size_t)ktn * 16 * 12;
        v2f n0 = *(const v2f*)(gn + 0);
        v2f n1 = *(const v2f*)(gn + 4);
        v2f n2 = *(const v2f*)(gn + 8);

        for (int t = 0; t < NPT; ++t) {
            v2f b0 = {bv[t][0], bv[t][1]};
            v2f b1 = {bv[t][2], bv[t][3]};
            v2f b2 = {bv[t][4], bv[t][5]};
            v8f d = {0.f, 0.f, 0.f, 0.f, 0.f, 0.f, 0.f, 0.f};
            d = __builtin_amdgcn_wmma_f32_16x16x4_f32(false, a0, false, b0,
                                                      (short)0, d, false, false);
            d = __builtin_amdgcn_wmma_f32_16x16x4_f32(false, a1, false, b1,
                                                      (short)0, d, false, false);
            d = __builtin_amdgcn_wmma_f32_16x16x4_f32(false, a2, false, b2,
                                                      (short)0, d, false, false);
            // 8 independent transcendentals, then a balanced tree reduction:
            // back-to-back v_exp_f32 fill each other's TRANS hazard slots.
            float e0 = __builtin_amdgcn_exp2f(d[0]);
            float e1 = __builtin_amdgcn_exp2f(d[1]);
            float e2 = __builtin_amdgcn_exp2f(d[2]);
            float e3 = __builtin_amdgcn_exp2f(d[3]);
            float e4 = __builtin_amdgcn_exp2f(d[4]);
            float e5 = __builtin_amdgcn_exp2f(d[5]);
            float e6 = __builtin_amdgcn_exp2f(d[6]);
            float e7 = __builtin_amdgcn_exp2f(d[7]);
            float s01 = e0 + e1;
            float s23 = e2 + e3;
            float s45 = e4 + e5;
            float s67 = e6 + e7;
            float s03 = s01 + s23;
            float s47 = s45 + s67;
            acc[t] += s03 + s47;
        }

        a0 = n0; a1 = n1; a2 = n2;
    }

    // lane L has gaussians {8h..8h+7} of each tile summed for point col;
    // partner lane L^16 has the other 8 rows.
    for (int t = 0; t < NPT; ++t) {
        float tot = acc[t] + __shfl_xor(acc[t], 16, 32);
        if (h == 0 && pidx[t] < N) out[pidx[t]] = tot;
    }
}

// ---------------------------------------------------------------------------
extern "C" void kernel_launch(void* const* d_in, const int* in_sizes, int n_in,
                              void* d_out, int out_size, void* d_ws, size_t ws_size,
                              hipStream_t stream) {
    const float* x          = (const float*)d_in[0];
    const float* means      = (const float*)d_in[1];
    const float* log_scales = (const float*)d_in[2];
    const float* quats      = (const float*)d_in[3];
    const float* log_amps   = (const float*)d_in[4];
    float* out = (float*)d_out;

    int N = in_sizes[0] / 3;  // points
    int K = in_sizes[4];      // gaussians (log_amplitudes is (K,))

    float* g = (float*)d_ws;  // K * 12 floats of fused coefficients

    gmf_precompute<<<(K + 255) / 256, 256, 0, stream>>>(means, log_scales, quats,
                                                        log_amps, g, K);

    int pts_per_block = 16 * NPT * WAVES_PER_BLOCK;  // 512
    int blocks = (N + pts_per_block - 1) / pts_per_block;
    gmf_eval<<<blocks, 32 * WAVES_PER_BLOCK, 0, stream>>>(x, g, out, N, K);
}